// SPR_GNN_88648124990705
// MI455X (gfx1250) — compile-verified
//
#include <hip/hip_runtime.h>
#include <hip/hip_bf16.h>

typedef __attribute__((ext_vector_type(2))) float v2f;
typedef __attribute__((ext_vector_type(8))) float v8f;

#define HID 64

// ---------------------------------------------------------------------------
// h[i][:] = node_emb[x[i]][:]   (float4 per thread, 16 threads per node row)
// ---------------------------------------------------------------------------
__global__ __launch_bounds__(256) void embed_kernel(
    const float* __restrict__ node_emb, const int* __restrict__ x,
    float* __restrict__ h, long long N)
{
    long long gid = (long long)blockIdx.x * blockDim.x + threadIdx.x;
    long long node = gid >> 4;
    int c4 = (int)(gid & 15);
    if (node >= N) return;
    long long v = (long long)x[node];
    float4 val = ((const float4*)(node_emb + v * HID))[c4];
    ((float4*)(h + node * HID))[c4] = val;
}

// ---------------------------------------------------------------------------
// Edge message + scatter-add:  aggr[dst] += relu(h[src] + (attr @ We.T + be))
// 16 threads per edge, 4 features each. Edge encoder recomputed on the fly
// (192 MACs) — far cheaper than storing an 819 MB e tensor.
// Atomic is skipped when the relu'd message is <= 0 (adding 0 is a no-op).
// h / aggr tables are L2-resident (25.6 MB each, L2 = 192 MB).
// ---------------------------------------------------------------------------
__global__ __launch_bounds__(256) void msg_kernel(
    const float* __restrict__ h, const int* __restrict__ edge_index,
    const float* __restrict__ edge_attr,
    const float* __restrict__ We, const float* __restrict__ be,
    float* __restrict__ aggr, long long E)
{
    long long gid = (long long)blockIdx.x * blockDim.x + threadIdx.x;
    long long edge = gid >> 4;
    int sub = (int)(gid & 15);
    if (edge >= E) return;

    int src = edge_index[edge];
    int dst = edge_index[E + edge];
    float a0 = edge_attr[edge * 3 + 0];
    float a1 = edge_attr[edge * 3 + 1];
    float a2 = edge_attr[edge * 3 + 2];

    float4 hv = ((const float4*)(h + (long long)src * HID))[sub];
    float hq[4] = {hv.x, hv.y, hv.z, hv.w};
    float* dstp = aggr + (long long)dst * HID + sub * 4;

#pragma unroll
    for (int q = 0; q < 4; ++q) {
        int k = sub * 4 + q;
        float e = be[k] + We[k * 3 + 0] * a0 + We[k * 3 + 1] * a1 + We[k * 3 + 2] * a2;
        float m = hq[q] + e;
        if (m > 0.0f) atomicAdd(dstp + q, m);   // global_atomic_add_f32
    }
}

// ---------------------------------------------------------------------------
// Fused GINE node MLP:  out = relu( relu((h+aggr) @ W1^T + b1) @ W2^T + b2 )
// One wave per 16-node tile, fp32 WMMA 16x16x4. 8 waves / 256-thread block.
// A-frag (16x4 f32): lane L -> row L%16, {K = 2*(L/16)+4*ks, +1}
// B-frag = W^T     : lane L -> col L%16, contiguous float2 of the weight row
// D       (16x16)  : acc[r] = D[r + 8*(L/16)][L%16]
// ---------------------------------------------------------------------------
__global__ __launch_bounds__(256) void mlp_kernel(
    const float* __restrict__ h, const float* __restrict__ aggr,
    const float* __restrict__ W1, const float* __restrict__ b1,
    const float* __restrict__ W2, const float* __restrict__ b2,
    float* __restrict__ out, long long N)
{
    __shared__ float sW1[HID * HID];
    __shared__ float sW2[HID * HID];
    __shared__ float sB1[HID];
    __shared__ float sB2[HID];
    __shared__ float sHA[8][16 * HID];   // per-wave (h + aggr) tile
    __shared__ float sZ[8][16 * HID];    // per-wave hidden tile

    int tid = threadIdx.x;

    // Stage weights into LDS (4096 floats each -> 1024 float4)
    for (int i = tid; i < (HID * HID) / 4; i += 256) {
        ((float4*)sW1)[i] = ((const float4*)W1)[i];
        ((float4*)sW2)[i] = ((const float4*)W2)[i];
    }
    if (tid < HID) { sB1[tid] = b1[tid]; sB2[tid] = b2[tid]; }
    __syncthreads();

    int wave = tid >> 5;            // wave32
    int lane = tid & 31;
    long long tile0 = (long long)blockIdx.x * 128 + (long long)wave * 16;

    // Load HA = h + aggr  (16 rows x 64 cols; 16 lanes cover one 256B row)
    float* ha = sHA[wave];
    for (int i = lane; i < 16 * 16; i += 32) {
        int row = i >> 4;
        int c4  = i & 15;
        long long node = tile0 + row;
        float4 v = make_float4(0.f, 0.f, 0.f, 0.f);
        if (node < N) {
            float4 a = ((const float4*)(h + node * HID))[c4];
            float4 g = ((const float4*)(aggr + node * HID))[c4];
            v = make_float4(a.x + g.x, a.y + g.y, a.z + g.z, a.w + g.w);
        }
        ((float4*)ha)[i] = v;
    }
    __syncthreads();

    int row   = lane & 15;          // A row / B-D column within tile
    int khalf = lane >> 4;          // 0 or 1
    float* z = sZ[wave];

    // GEMM1: Z = relu(HA @ W1^T + b1)
    for (int ct = 0; ct < 4; ++ct) {
        v8f acc = {};
        int ncol = ct * 16 + row;
#pragma unroll
        for (int ks = 0; ks < 16; ++ks) {
            int kb = ks * 4 + khalf * 2;
            v2f a = *(const v2f*)(ha + row * HID + kb);
            v2f b = *(const v2f*)(sW1 + ncol * HID + kb);
            acc = __builtin_amdgcn_wmma_f32_16x16x4_f32(
                false, a, false, b, (short)0, acc, false, false);
        }
        float bias = sB1[ncol];
#pragma unroll
        for (int r = 0; r < 8; ++r) {
            int m = r + khalf * 8;
            float v = acc[r] + bias;
            z[m * HID + ncol] = v > 0.f ? v : 0.f;
        }
    }
    __syncthreads();

    // GEMM2: OUT = relu(Z @ W2^T + b2)
    for (int ct = 0; ct < 4; ++ct) {
        v8f acc = {};
        int ncol = ct * 16 + row;
#pragma unroll
        for (int ks = 0; ks < 16; ++ks) {
            int kb = ks * 4 + khalf * 2;
            v2f a = *(const v2f*)(z + row * HID + kb);
            v2f b = *(const v2f*)(sW2 + ncol * HID + kb);
            acc = __builtin_amdgcn_wmma_f32_16x16x4_f32(
                false, a, false, b, (short)0, acc, false, false);
        }
        float bias = sB2[ncol];
#pragma unroll
        for (int r = 0; r < 8; ++r) {
            int m = r + khalf * 8;
            long long node = tile0 + m;
            if (node < N) {
                float v = acc[r] + bias;
                out[node * HID + ncol] = v > 0.f ? v : 0.f;
            }
        }
    }
}

// ---------------------------------------------------------------------------
// Segment-max pooling. Final h >= 0 (outer relu), so IEEE bit patterns are
// monotone as unsigned ints -> atomicMax on uint bits; pooled pre-zeroed,
// which also implements the "-inf -> 0 for empty segments" guard exactly.
// ---------------------------------------------------------------------------
__global__ __launch_bounds__(256) void pool_kernel(
    const float* __restrict__ h, const int* __restrict__ batch,
    unsigned int* __restrict__ pooled_bits, long long N)
{
    long long gid = (long long)blockIdx.x * blockDim.x + threadIdx.x;
    long long node = gid >> 4;
    int c4 = (int)(gid & 15);
    if (node >= N) return;
    int g = batch[node];
    float4 v = ((const float4*)(h + node * HID))[c4];
    unsigned int* p = pooled_bits + (long long)g * HID + c4 * 4;
    atomicMax(p + 0, __float_as_uint(v.x));
    atomicMax(p + 1, __float_as_uint(v.y));
    atomicMax(p + 2, __float_as_uint(v.z));
    atomicMax(p + 3, __float_as_uint(v.w));
}

// ---------------------------------------------------------------------------
// Classifier head: out[g][c] = pooled[g] . Wc[c] + bc[c]   (G*C = 5120 dots)
// ---------------------------------------------------------------------------
__global__ __launch_bounds__(256) void head_kernel(
    const float* __restrict__ pooled, const float* __restrict__ Wc,
    const float* __restrict__ bc, float* __restrict__ out, int G, int C)
{
    int gid = blockIdx.x * blockDim.x + threadIdx.x;
    if (gid >= G * C) return;
    int g = gid / C;
    int c = gid - g * C;
    const float* p = pooled + (long long)g * HID;
    const float* w = Wc + (long long)c * HID;
    float s = bc[c];
#pragma unroll
    for (int k = 0; k < HID; ++k) s += p[k] * w[k];
    out[gid] = s;
}

// ---------------------------------------------------------------------------
extern "C" void kernel_launch(void* const* d_in, const int* in_sizes, int n_in,
                              void* d_out, int out_size, void* d_ws, size_t ws_size,
                              hipStream_t stream)
{
    const int*   x          = (const int*)  d_in[0];
    const int*   edge_index = (const int*)  d_in[1];
    const float* edge_attr  = (const float*)d_in[2];
    const int*   batch      = (const int*)  d_in[3];
    // d_in[4] = num_graphs (device scalar) — derived on host instead
    const float* node_emb   = (const float*)d_in[5];
    const float* We  = (const float*)d_in[6];
    const float* be  = (const float*)d_in[7];
    const float* W1a = (const float*)d_in[8];
    const float* b1a = (const float*)d_in[9];
    const float* W2a = (const float*)d_in[10];
    const float* b2a = (const float*)d_in[11];
    const float* W1b = (const float*)d_in[12];
    const float* b1b = (const float*)d_in[13];
    const float* W2b = (const float*)d_in[14];
    const float* b2b = (const float*)d_in[15];
    const float* Wc  = (const float*)d_in[16];
    const float* bc  = (const float*)d_in[17];

    const long long N = in_sizes[0];
    const long long E = (long long)in_sizes[1] / 2;
    const int C = in_sizes[17];          // bc length = n_classes
    const int G = out_size / C;

    // Workspace: two N x 64 node buffers (ping/pong) + pooled
    float* hA     = (float*)d_ws;
    float* hB     = hA + N * HID;
    float* pooled = hB + N * HID;

    const int TPB = 256;
    dim3 blk(TPB);
    dim3 grid_node((unsigned)((N * 16 + TPB - 1) / TPB));
    dim3 grid_edge((unsigned)((E * 16 + TPB - 1) / TPB));
    dim3 grid_mlp((unsigned)((N + 127) / 128));
    dim3 grid_head((unsigned)((G * C + TPB - 1) / TPB));

    // ---- Layer A ----
    hipMemsetAsync(hB, 0, (size_t)N * HID * sizeof(float), stream);   // aggr
    embed_kernel<<<grid_node, blk, 0, stream>>>(node_emb, x, hA, N);
    msg_kernel<<<grid_edge, blk, 0, stream>>>(hA, edge_index, edge_attr, We, be, hB, E);
    // reads h=hA, aggr=hB; writes new h in place over hB (per-tile RAW-safe)
    mlp_kernel<<<grid_mlp, blk, 0, stream>>>(hA, hB, W1a, b1a, W2a, b2a, hB, N);

    // ---- Layer B ----
    hipMemsetAsync(hA, 0, (size_t)N * HID * sizeof(float), stream);   // aggr
    msg_kernel<<<grid_edge, blk, 0, stream>>>(hB, edge_index, edge_attr, We, be, hA, E);
    mlp_kernel<<<grid_mlp, blk, 0, stream>>>(hB, hA, W1b, b1b, W2b, b2b, hA, N);

    // ---- Pool + head ----
    hipMemsetAsync(pooled, 0, (size_t)G * HID * sizeof(float), stream);
    pool_kernel<<<grid_node, blk, 0, stream>>>(hA, batch, (unsigned int*)pooled, N);
    head_kernel<<<grid_head, blk, 0, stream>>>(pooled, Wc, bc, (float*)d_out, G, C);
}